// Net_67886253080950
// MI455X (gfx1250) — compile-verified
//
#include <hip/hip_runtime.h>
#include <hip/hip_bf16.h>

typedef __attribute__((ext_vector_type(2))) float v2f;
typedef __attribute__((ext_vector_type(4))) float v4f;
typedef __attribute__((ext_vector_type(8))) float v8f;

#define B_      32
#define NUM_EXO 8
#define HW      784
#define C_      512
#define H_      28
#define EPS_    1e-12f

#define N_EXO_ROWS (B_ * NUM_EXO * HW)   // 200704
#define N_EGO_ROWS (B_ * HW)             // 25088
#define NT_EXO     (N_EXO_ROWS / 16)     // 12544
#define NT_EGO     (N_EGO_ROWS / 16)     // 1568
#define NT_TOTAL   (NT_EXO + NT_EGO)     // 14112
#define WAVES_PER_BLOCK 8

// ---------------------------------------------------------------------------
// Kernel A: inverse L2 norms of the 32 aff rows and 32 txt rows.
// 64 blocks x 32 threads (one wave per vector).
// ---------------------------------------------------------------------------
__global__ void vecnorm_kernel(const float* __restrict__ aff,
                               const float* __restrict__ txt,
                               float* __restrict__ inv_aff,
                               float* __restrict__ inv_txt) {
    int bidx = blockIdx.x;
    int lane = threadIdx.x;
    const float* src;
    float* dst;
    int r;
    if (bidx < B_) { src = aff; dst = inv_aff; r = bidx; }
    else           { src = txt; dst = inv_txt; r = bidx - B_; }
    const float* p = src + (size_t)r * C_;
    float s = 0.f;
    for (int k = lane; k < C_; k += 32) { float v = p[k]; s += v * v; }
#pragma unroll
    for (int off = 16; off > 0; off >>= 1) s += __shfl_xor(s, off, 32);
    if (lane == 0) dst[r] = 1.f / fmaxf(sqrtf(s), EPS_);
}

// ---------------------------------------------------------------------------
// Kernel B: per-16-row tile, WMMA f32 16x16x4 skinny GEMM:
//   D col0 = row . primary vector (txt for exo, aff for ego)
//   D col1 = row . aff            (exo only; garbage for ego, discarded)
//   D cols 2..15 = garbage (finite), never read.
// K-slots are permuted (lane group kg covers global k = kb + 4*kg + 0..3) so
// each lane does ONE contiguous b128 load per 8-K chunk. Dot products are
// K-order invariant, and A/B use the identical permutation.
// Row norms (row.row) accumulate per-lane from the same A data.
// ---------------------------------------------------------------------------
__global__ __launch_bounds__(32 * WAVES_PER_BLOCK)
void dots_kernel(const float* __restrict__ exo,
                 const float* __restrict__ ego,
                 const float* __restrict__ aff,
                 const float* __restrict__ txt,
                 const float* __restrict__ inv_aff,
                 const float* __restrict__ inv_txt,
                 float* __restrict__ sim_obj,   // ws, N_EXO_ROWS
                 float* __restrict__ sim_aff,   // ws, N_EXO_ROWS
                 float* __restrict__ out_ego) { // d_out, N_EGO_ROWS
    __shared__ float s_obj[WAVES_PER_BLOCK][16];
    __shared__ float s_aff[WAVES_PER_BLOCK][16];
    __shared__ float s_inv[WAVES_PER_BLOCK][16];

    const int wslot = threadIdx.x >> 5;
    const int lane  = threadIdx.x & 31;
    const int m  = lane & 15;   // row within tile (A) / column n (B/D)
    const int kg = lane >> 4;   // K-group (which half of each 8-K chunk)

    const int wave = blockIdx.x * WAVES_PER_BLOCK + wslot;
    const bool is_exo = (wave < NT_EXO);

    int row0, batch;
    const float* arow;
    const float* v0;
    const float* v1;
    if (is_exo) {
        row0  = wave * 16;
        batch = row0 / (NUM_EXO * HW);
        arow  = exo + (size_t)(row0 + m) * C_;
        v0    = txt + (size_t)batch * C_;
        v1    = aff + (size_t)batch * C_;
    } else {
        row0  = (wave - NT_EXO) * 16;
        batch = row0 / HW;
        arow  = ego + (size_t)(row0 + m) * C_;
        v0    = aff + (size_t)batch * C_;
        v1    = v0;
    }
    arow += 4 * kg;
    // col0 lanes read the primary vector; every other lane reads aff (cols
    // >= 2 produce garbage that is never read -- no masking, no exec branch).
    const float* bvec = ((m == 0) ? v0 : v1) + 4 * kg;

    v8f acc0 = {}, acc1 = {}, acc2 = {}, acc3 = {};
    float ns = 0.f;

    for (int kb = 0; kb < C_; kb += 16) {
        // A streams through once (411 MB >> L2): non-temporal.
        v4f ax = __builtin_nontemporal_load((const v4f*)(arow + kb));
        v4f ay = __builtin_nontemporal_load((const v4f*)(arow + kb + 8));
        v4f bx = *(const v4f*)(bvec + kb);
        v4f by = *(const v4f*)(bvec + kb + 8);

        ns += ax[0] * ax[0] + ax[1] * ax[1] + ax[2] * ax[2] + ax[3] * ax[3];
        ns += ay[0] * ay[0] + ay[1] * ay[1] + ay[2] * ay[2] + ay[3] * ay[3];

        v2f a01 = {ax[0], ax[1]}, a23 = {ax[2], ax[3]};
        v2f b01 = {bx[0], bx[1]}, b23 = {bx[2], bx[3]};
        v2f c01 = {ay[0], ay[1]}, c23 = {ay[2], ay[3]};
        v2f d01 = {by[0], by[1]}, d23 = {by[2], by[3]};

        acc0 = __builtin_amdgcn_wmma_f32_16x16x4_f32(false, a01, false, b01,
                                                     (short)0, acc0, false, false);
        acc1 = __builtin_amdgcn_wmma_f32_16x16x4_f32(false, a23, false, b23,
                                                     (short)0, acc1, false, false);
        acc2 = __builtin_amdgcn_wmma_f32_16x16x4_f32(false, c01, false, d01,
                                                     (short)0, acc2, false, false);
        acc3 = __builtin_amdgcn_wmma_f32_16x16x4_f32(false, c23, false, d23,
                                                     (short)0, acc3, false, false);
    }

    v8f acc;
#pragma unroll
    for (int i = 0; i < 8; ++i) acc[i] = (acc0[i] + acc1[i]) + (acc2[i] + acc3[i]);

    // full row norm^2: combine the two K-halves (lane m <-> lane m+16)
    float nsfull = ns + __shfl_xor(ns, 16, 32);
    float invrow = 1.f / fmaxf(sqrtf(nsfull), EPS_);

    // D layout: element i of lane (n=m, kg) is D[row = i + 8*kg][col = m]
    if (m == 0) {
#pragma unroll
        for (int i = 0; i < 8; ++i) s_obj[wslot][8 * kg + i] = acc[i];
    }
    if (m == 1) {
#pragma unroll
        for (int i = 0; i < 8; ++i) s_aff[wslot][8 * kg + i] = acc[i];
    }
    if (lane < 16) s_inv[wslot][lane] = invrow;

    __syncthreads();

    if (lane < 16) {
        int row = row0 + lane;
        float io = s_inv[wslot][lane];
        if (is_exo) {
            sim_obj[row] = s_obj[wslot][lane] * io * inv_txt[batch];
            sim_aff[row] = s_aff[wslot][lane] * io * inv_aff[batch];
        } else {
            out_ego[row] = s_obj[wslot][lane] * io * inv_aff[batch];
        }
    }
}

// ---------------------------------------------------------------------------
// Kernel C: edge-aware 3x3 box mean of sim_obj, multiplied by sim_aff.
// ---------------------------------------------------------------------------
__global__ void filter_kernel(const float* __restrict__ sim_obj,
                              const float* __restrict__ sim_aff,
                              float* __restrict__ out_exo) {
    int idx = blockIdx.x * blockDim.x + threadIdx.x;
    if (idx >= N_EXO_ROWS) return;
    int x   = idx % H_;
    int y   = (idx / H_) % H_;
    int img = idx / HW;
    const float* so = sim_obj + (size_t)img * HW;
    float s = 0.f;
    float cnt = 0.f;
#pragma unroll
    for (int dy = -1; dy <= 1; ++dy) {
        int yy = y + dy;
        if (yy < 0 || yy >= H_) continue;
#pragma unroll
        for (int dx = -1; dx <= 1; ++dx) {
            int xx = x + dx;
            if (xx < 0 || xx >= H_) continue;
            s += so[yy * H_ + xx];
            cnt += 1.f;
        }
    }
    out_exo[idx] = (s / cnt) * sim_aff[idx];
}

// ---------------------------------------------------------------------------
extern "C" void kernel_launch(void* const* d_in, const int* in_sizes, int n_in,
                              void* d_out, int out_size, void* d_ws, size_t ws_size,
                              hipStream_t stream) {
    const float* exo = (const float*)d_in[0];   // (B*NUM_EXO, HW, C)
    const float* ego = (const float*)d_in[1];   // (B, HW, C)
    const float* aff = (const float*)d_in[2];   // (B, C)
    const float* txt = (const float*)d_in[3];   // (B, C)

    float* out = (float*)d_out;                 // [ego (25088)] ++ [exo (200704)]
    float* out_ego = out;
    float* out_exo = out + N_EGO_ROWS;

    float* wsf      = (float*)d_ws;
    float* inv_aff  = wsf;                      // 32
    float* inv_txt  = wsf + 32;                 // 32
    float* sim_obj  = wsf + 64;                 // 200704
    float* sim_aff  = sim_obj + N_EXO_ROWS;     // 200704

    vecnorm_kernel<<<2 * B_, 32, 0, stream>>>(aff, txt, inv_aff, inv_txt);

    dots_kernel<<<NT_TOTAL / WAVES_PER_BLOCK, 32 * WAVES_PER_BLOCK, 0, stream>>>(
        exo, ego, aff, txt, inv_aff, inv_txt, sim_obj, sim_aff, out_ego);

    filter_kernel<<<(N_EXO_ROWS + 255) / 256, 256, 0, stream>>>(
        sim_obj, sim_aff, out_exo);
}